// KHReverbDSP_54571854463261
// MI455X (gfx1250) — compile-verified
//
#include <hip/hip_runtime.h>
#include <stdint.h>

#define SRATE   44100
#define NBATCH  64
#define NSAMP   44100
#define NDEL    8
#define MAXBUF  7266
#define CHUNK   256

__device__ __constant__ float c_base[NDEL] =
    {487.f, 601.f, 719.f, 853.f, 1013.f, 1153.f, 1297.f, 1453.f};

// ---------------------------------------------------------------------------
// Stage 1: sequential FDN scan. One wave32 per batch; lanes 0..7 = delay lines.
// Writes lateL -> out[b][0][t], lateR -> out[b][1][t].
// Delay-line circular buffers (length d per line) live in global scratch
// (14.9 MB total -> L2 resident). Delay reads are software-pipelined one
// 8-step group ahead (legal: min delay 243 >> 16). Input x is staged into LDS
// with CDNA5 async global->LDS copies, double buffered per 256-sample chunk.
// ---------------------------------------------------------------------------
__global__ __launch_bounds__(32)
void KHReverb_fdn_scan(const float* __restrict__ x,
                       const float* __restrict__ decay,
                       const float* __restrict__ dampen,
                       const float* __restrict__ size_p,
                       float* __restrict__ out,
                       float* __restrict__ bufws)
{
    const int b    = blockIdx.x;
    const int lane = threadIdx.x;
    const int l    = lane & 7;

    __shared__ float xs[2][CHUNK];

    // per-batch scalar params (match reference math, fp32)
    const float scale0 = 0.5f + size_p[0] * 4.5f;          // batch-0 size -> delay lengths
    const float scaleb = 0.5f + size_p[b] * 4.5f;          // this batch's size -> feedback gains
    const float t60    = 0.1f + decay[b] * 9.9f;
    const float cutoff = 500.f + (1.f - dampen[b]) * 19500.f;
    const float alpha  = expf(-6.2831853071795865f * cutoff / (float)SRATE);
    const float one_m_alpha = 1.f - alpha;

    const float base_l = c_base[l];
    const int   d      = (int)floorf(base_l * scale0);     // this line's delay (samples)
    const float g      = powf(10.f, -3.f * (base_l * scaleb / (float)SRATE) / t60);

    float* __restrict__ buf  = bufws + ((size_t)b * NDEL + l) * MAXBUF;
    const float* __restrict__ xb = x + (size_t)b * NSAMP;
    float* __restrict__ outL = out + (size_t)b * 2 * NSAMP;
    float* __restrict__ outR = outL + NSAMP;

    float filt = 0.f;

    // ---- async staging of x into LDS --------------------------------------
    auto prefetch = [&](int c, int sel) {
        const int start = c * CHUNK;
        int k = lane;
#pragma unroll
        for (int it = 0; it < CHUNK / 32; ++it, k += 32) {
            const int tg = start + k;
            if (tg < NSAMP) {
                uint32_t lds_a = (uint32_t)(uintptr_t)(&xs[sel][k]); // low 32b = LDS offset
                const float* gp = xb + tg;
                asm volatile("global_load_async_to_lds_b32 %0, %1, off"
                             :: "v"(lds_a), "v"(gp) : "memory");
            }
        }
    };

    // ---- pipelined delay-line group loads ---------------------------------
    // loads the 8 delayed samples for steps [tl, tl+8) at circular index base idxl
    auto load_group = [&](float (&dst)[8], int tl, int idxl) {
#pragma unroll
        for (int j = 0; j < 8; ++j) {
            dst[j] = 0.f;
            const int t = tl + j;
            if (lane < 8 && t < NSAMP && t >= d) {
                int id2 = idxl + j; if (id2 >= d) id2 -= d;
                dst[j] = buf[id2];
            }
        }
    };

    float del_cur[8], del_nxt[8];
    int tl = 0, idxl = 0;      // load pointer (group being fetched)
    int idxc = 0;              // compute pointer: t mod d of current group start

    load_group(del_cur, tl, idxl);
    tl += 8; idxl += 8; if (idxl >= d) idxl -= d;

    const int nchunks = (NSAMP + CHUNK - 1) / CHUNK;
    prefetch(0, 0);

    for (int c = 0; c < nchunks; ++c) {
        // copy issued one full chunk (256 steps) earlier -> wait is ~free
        asm volatile("s_wait_asynccnt 0x0" ::: "memory");
        if (c + 1 < nchunks) prefetch(c + 1, (c + 1) & 1);

        const float* __restrict__ xc = xs[c & 1];
        const int t0     = c * CHUNK;
        const int nsteps = (NSAMP - t0 < CHUNK) ? (NSAMP - t0) : CHUNK;

        for (int j0 = 0; j0 < nsteps; j0 += 8) {
            const int m = (nsteps - j0 < 8) ? (nsteps - j0) : 8;

            // issue next group's delay reads now; they complete while we
            // compute the current 8 steps (L2 latency fully overlapped)
            if (tl < NSAMP) load_group(del_nxt, tl, idxl);
            tl += 8; idxl += 8; if (idxl >= d) idxl -= d;

#pragma unroll
            for (int j = 0; j < 8; ++j) {
                if (j < m) {                                // wave-uniform predicate
                    const int t  = t0 + j0 + j;
                    const float xt = xc[t - t0];

                    // one-pole LPF
                    const float fi = one_m_alpha * del_cur[j] + alpha * filt;
                    filt = fi;

                    // Householder mix: mixed_i = v_i - 0.25 * sum_j v_j
                    float v = fi * g;
                    float s = v;
                    s += __shfl_xor(s, 1);
                    s += __shfl_xor(s, 2);
                    s += __shfl_xor(s, 4);
                    const float w = xt + v - 0.25f * s;

                    if (lane < 8) {
                        int id2 = idxc + j; if (id2 >= d) id2 -= d;
                        buf[id2] = w;                       // feedback write
                    }

                    // outL = sum filt[0:4], outR = sum filt[4:8]
                    float h = fi;
                    h += __shfl_xor(h, 1);
                    h += __shfl_xor(h, 2);                  // half-sums within groups of 4
                    const float ho = __shfl_xor(h, 4);
                    if (lane == 0) { outL[t] = h; outR[t] = ho; }
                }
            }
            idxc += m; if (idxc >= d) idxc -= d;

#pragma unroll
            for (int j = 0; j < 8; ++j) del_cur[j] = del_nxt[j];
        }
    }
}

// ---------------------------------------------------------------------------
// Stage 2: early reflections + mid/side width + dry/wet mix. Elementwise over
// (b, t); reads lateL/lateR from d_out (written by stage 1) and RMWs in place.
// ---------------------------------------------------------------------------
__global__ __launch_bounds__(256)
void KHReverb_post(const float* __restrict__ x,
                   const float* __restrict__ size_p,
                   const float* __restrict__ width_p,
                   const float* __restrict__ early_p,
                   const float* __restrict__ mix_p,
                   float* __restrict__ out)
{
    const int gid = blockIdx.x * blockDim.x + threadIdx.x;
    if (gid >= NBATCH * NSAMP) return;
    const int b = gid / NSAMP;
    const int t = gid - b * NSAMP;

    const float scale0 = 0.5f + size_p[0] * 4.5f;
    const float wdt  = width_p[b];
    const float eg   = early_p[b] * 0.5f;
    const float mix  = mix_p[b];

    const float* __restrict__ xb = x + (size_t)b * NSAMP;
    const float xt = xb[t];

    float eL = 0.f, eR = 0.f;
    float gain = eg;
#pragma unroll
    for (int i = 0; i < 4; ++i) {
        const int tap = (int)floorf(100.f * (float)(i + 1) * scale0);
        const float ta = (t >= tap && tap < NSAMP) ? xb[t - tap] : 0.f;
        eL += ta * gain;
        eR += ta * gain * (1.f - (float)(i & 1) * wdt);
        gain *= 0.8f;
    }

    float* pL = out + (size_t)b * 2 * NSAMP + t;
    float* pR = pL + NSAMP;
    float lL = *pL, lR = *pR;                // late reflections from stage 1
    const float mid  = (lL + lR) * 0.5f;
    const float side = (lL - lR) * 0.5f;
    lL = mid + side * wdt;
    lR = mid - side * wdt;

    *pL = (1.f - mix) * xt + mix * (eL + lL);
    *pR = (1.f - mix) * xt + mix * (eR + lR);
}

// ---------------------------------------------------------------------------
extern "C" void kernel_launch(void* const* d_in, const int* in_sizes, int n_in,
                              void* d_out, int out_size, void* d_ws, size_t ws_size,
                              hipStream_t stream)
{
    // setup_inputs order: x, decay, dampen, size, width, early, mix
    const float* x      = (const float*)d_in[0];
    const float* decay  = (const float*)d_in[1];
    const float* dampen = (const float*)d_in[2];
    const float* size_p = (const float*)d_in[3];
    const float* width  = (const float*)d_in[4];
    const float* early  = (const float*)d_in[5];
    const float* mix    = (const float*)d_in[6];
    float* out = (float*)d_out;

    // scratch: delay-line state, 64 * 8 * 7266 * 4B ~= 14.9 MB (fits L2)
    float* bufws = (float*)d_ws;

    KHReverb_fdn_scan<<<dim3(NBATCH), dim3(32), 0, stream>>>(
        x, decay, dampen, size_p, out, bufws);

    const int total = NBATCH * NSAMP;
    KHReverb_post<<<dim3((total + 255) / 256), dim3(256), 0, stream>>>(
        x, size_p, width, early, mix, out);
}